// AffineNet_71081708749088
// MI455X (gfx1250) — compile-verified
//
#include <hip/hip_runtime.h>

typedef float v2f __attribute__((ext_vector_type(2)));
typedef float v8f __attribute__((ext_vector_type(8)));

#define B_ 8
#define C_ 8
#define H_ 256
#define W_ 256
#define O_ 8

// 8-byte gather with only 4-byte alignment guarantee (xp is arbitrary):
// let the compiler pick b64-unaligned or 2xb32.
__device__ __forceinline__ float2 ld2(const float* p) {
    float2 r;
    __builtin_memcpy(&r, p, 8);
    return r;
}

// Bilinear sample of one channel with pre-combined (weight*validity) factors.
// Indices are pre-clamped so every load is in-bounds; invalid corners carry
// zero weight, matching the reference's zeros padding.
__device__ __forceinline__ float sample_ch(const float* __restrict__ ch,
                                           int y0, int y1, int xp,
                                           bool sx0, bool sx1,
                                           float wA, float wB, float wC, float wD) {
    float2 p0 = ld2(ch + y0 * W_ + xp);
    float2 p1 = ld2(ch + y1 * W_ + xp);
    float v00 = sx0 ? p0.y : p0.x;
    float v01 = sx1 ? p0.y : p0.x;
    float v10 = sx0 ? p1.y : p1.x;
    float v11 = sx1 ? p1.y : p1.x;
    return wA * v00 + wB * v01 + wC * v10 + wD * v11;
}

__global__ __launch_bounds__(256) void affinenet_wmma_kernel(
        const float* __restrict__ x,      // [B,C,H,W]
        const float* __restrict__ theta,  // [O,2,3]
        float* __restrict__ out)          // [B,O,H,W]
{
    const int bo = blockIdx.y;            // 0..63 = b*8 + o (block-uniform)
    const int b  = bo >> 3;
    const int o  = bo & 7;

    const float* th = theta + o * 6;
    const float t00 = th[0], t01 = th[1], t02 = th[2];
    const float t10 = th[3], t11 = th[4], t12 = th[5];

    const int lane = threadIdx.x & 31;
    const int wave = threadIdx.x >> 5;    // 0..7
    const int px   = lane & 15;           // pixel within 16-wide strip
    const int chb  = (lane >> 4) << 1;    // 0 for lanes 0-15 (K=0,1), 2 for lanes 16-31 (K=2,3)

    const float* xb = x + (size_t)b * C_ * H_ * W_;
    float*       ob = out + (size_t)bo * H_ * W_;

    v2f bmat = {0.125f, 0.125f};          // uniform B: mean over C=8 via 2 WMMAs

    // 4096 strips per (b,o) image; gridDim.x*8 waves; exactly 8 strips/wave ->
    // loop trip count is wave-uniform so EXEC is all-ones at every WMMA.
    const int wavesTotal = gridDim.x * 8;
    for (int strip = blockIdx.x * 8 + wave; strip < H_ * (W_ / 16); strip += wavesTotal) {
        const int h     = strip >> 4;
        const int wbase = (strip & 15) << 4;
        const int wpix  = wbase + px;

        // affine grid, align_corners=True
        float fx = wpix * (2.0f / 255.0f) - 1.0f;
        float fy = h    * (2.0f / 255.0f) - 1.0f;
        float gx = t00 * fx + t01 * fy + t02;
        float gy = t10 * fx + t11 * fy + t12;
        float ix = (gx + 1.0f) * (0.5f * (W_ - 1));
        float iy = (gy + 1.0f) * (0.5f * (H_ - 1));

        float ix0f = floorf(ix), iy0f = floorf(iy);
        float ix1f = ix0f + 1.0f, iy1f = iy0f + 1.0f;
        float wx1 = ix - ix0f, wx0 = 1.0f - wx1;
        float wy1 = iy - iy0f, wy0 = 1.0f - wy1;

        float vx0 = (ix0f >= 0.0f && ix0f <= (float)(W_ - 1)) ? 1.0f : 0.0f;
        float vx1 = (ix1f >= 0.0f && ix1f <= (float)(W_ - 1)) ? 1.0f : 0.0f;
        float vy0 = (iy0f >= 0.0f && iy0f <= (float)(H_ - 1)) ? 1.0f : 0.0f;
        float vy1 = (iy1f >= 0.0f && iy1f <= (float)(H_ - 1)) ? 1.0f : 0.0f;

        float wA = wy0 * vy0 * wx0 * vx0;   // (y0,x0)
        float wB = wy0 * vy0 * wx1 * vx1;   // (y0,x1)
        float wC = wy1 * vy1 * wx0 * vx0;   // (y1,x0)
        float wD = wy1 * vy1 * wx1 * vx1;   // (y1,x1)

        // clamped, always-in-bounds gather indices (xp in [0, W-2] so the
        // float2 never crosses the buffer end)
        int xp = (int)fminf(fmaxf(ix0f, 0.0f), (float)(W_ - 2));
        int y0 = (int)fminf(fmaxf(iy0f, 0.0f), (float)(H_ - 1));
        int y1 = (int)fminf(fmaxf(iy1f, 0.0f), (float)(H_ - 1));
        bool sx0 = ix0f > (float)xp;        // pick .y half of the pair
        bool sx1 = ix1f > (float)xp;

        v8f acc = {};
        v2f a;
        // WMMA #1: channels 0..3 (this lane: chb, chb+1)
        a.x = sample_ch(xb + (size_t)(chb    ) * H_ * W_, y0, y1, xp, sx0, sx1, wA, wB, wC, wD);
        a.y = sample_ch(xb + (size_t)(chb + 1) * H_ * W_, y0, y1, xp, sx0, sx1, wA, wB, wC, wD);
        acc = __builtin_amdgcn_wmma_f32_16x16x4_f32(false, a, false, bmat, (short)0, acc, false, false);
        // WMMA #2: channels 4..7
        a.x = sample_ch(xb + (size_t)(chb + 4) * H_ * W_, y0, y1, xp, sx0, sx1, wA, wB, wC, wD);
        a.y = sample_ch(xb + (size_t)(chb + 5) * H_ * W_, y0, y1, xp, sx0, sx1, wA, wB, wC, wD);
        acc = __builtin_amdgcn_wmma_f32_16x16x4_f32(false, a, false, bmat, (short)0, acc, false, false);

        // D layout: VGPR r holds M=r (lanes 0-15) / M=8+r (lanes 16-31), all
        // columns identical. Lanes 0-7 write pixels 0-7, lanes 16-23 pixels 8-15.
        int k = lane & 7;
        float e0 = (k & 1) ? acc[1] : acc[0];
        float e1 = (k & 1) ? acc[3] : acc[2];
        float e2 = (k & 1) ? acc[5] : acc[4];
        float e3 = (k & 1) ? acc[7] : acc[6];
        float f0 = (k & 2) ? e1 : e0;
        float f1 = (k & 2) ? e3 : e2;
        float val = (k & 4) ? f1 : f0;
        if ((lane & 15) < 8) {
            int pixel = ((lane >> 4) << 3) + k;
            ob[h * W_ + wbase + pixel] = val;
        }
    }
}

extern "C" void kernel_launch(void* const* d_in, const int* in_sizes, int n_in,
                              void* d_out, int out_size, void* d_ws, size_t ws_size,
                              hipStream_t stream) {
    (void)in_sizes; (void)n_in; (void)out_size; (void)d_ws; (void)ws_size;
    const float* x     = (const float*)d_in[0];
    const float* theta = (const float*)d_in[1];
    float*       out   = (float*)d_out;

    // grid.y = 64 (b,o) pairs; grid.x*8 waves cover 4096 strips, 8 strips/wave.
    dim3 grid(64, 64, 1);
    affinenet_wmma_kernel<<<grid, 256, 0, stream>>>(x, theta, out);
}